// Freq2DPositionalEncoding_46342697124260
// MI455X (gfx1250) — compile-verified
//
#include <hip/hip_runtime.h>
#include <cstdint>

// out[b, f*T + t, e] = x[b, f*T+t, e] + pe[f, e] + pe[t, e]
// B=4, F=256, T=512, E=512, fp32.  Pure bandwidth problem: 2 GiB HBM traffic,
// ~92us floor at 23.3 TB/s.  pe table (1 MB) kept L2-resident by streaming
// x/out with NT hints; pe_f held in registers; pe_t prefetched through the
// gfx1250 async-to-LDS pipeline (ASYNCcnt), double buffered.

typedef float v4f __attribute__((ext_vector_type(4)));

#define B_      4
#define F_      256
#define T_      512
#define E_      512
#define THREADS 256
#define ITERS   (T_ / 2)   // 2 t-rows per iteration

__device__ __forceinline__ void async_copy_b128(const float* gsrc, void* lds_dst) {
    // VDST = LDS byte offset (low 32 bits of the generic pointer == LDS offset
    // per the flat->LDS aperture mapping), VADDR = 64-bit global address.
    unsigned loff = (unsigned)(uintptr_t)lds_dst;
    asm volatile("global_load_async_to_lds_b128 %0, %1, off"
                 :: "v"(loff), "v"(gsrc)
                 : "memory");
}

__global__ __launch_bounds__(THREADS) void pe2d_add_kernel(
    const float* __restrict__ x,
    const float* __restrict__ pe,
    float* __restrict__ out)
{
    __shared__ v4f pt_sh[2][THREADS];   // 8 KB double buffer for pe_t rows

    const int tid  = threadIdx.x;
    const int e4   = tid & 127;         // float4 column within a row (E/4 = 128)
    const int tsub = tid >> 7;          // which of the 2 t-rows this thread covers
    const int bf   = blockIdx.x;        // b*F + f, 0..1023
    const int f    = bf & (F_ - 1);

    // pe_f for this thread's column is loop-invariant -> registers.
    const v4f pf = *(const v4f*)(pe + (size_t)f * E_ + e4 * 4);

    const float* xrow = x   + (size_t)bf * T_ * E_;
    float*       orow = out + (size_t)bf * T_ * E_;

    // Prime the pipeline: stage t-rows {0,1}.
    async_copy_b128(pe + (size_t)tsub * E_ + e4 * 4, &pt_sh[0][tid]);

    for (int it = 0; it < ITERS; ++it) {
        if (it + 1 < ITERS) {
            // Prefetch next pair of t-rows into the other buffer.
            async_copy_b128(pe + (size_t)(2 * (it + 1) + tsub) * E_ + e4 * 4,
                            &pt_sh[(it + 1) & 1][tid]);
            // 2 outstanding, in-order completion -> <=1 means current buffer done.
            asm volatile("s_wait_asynccnt 1" ::: "memory");
        } else {
            asm volatile("s_wait_asynccnt 0" ::: "memory");
        }

        const v4f   pt  = pt_sh[it & 1][tid];
        const int   t   = 2 * it + tsub;
        const size_t off = (size_t)t * E_ + e4 * 4;

        // NT: don't let the 2 GiB stream evict the 1 MB pe table from L2.
        v4f xv = __builtin_nontemporal_load((const v4f*)(xrow + off));
        v4f o  = xv + pf + pt;
        __builtin_nontemporal_store(o, (v4f*)(orow + off));
    }
}

extern "C" void kernel_launch(void* const* d_in, const int* in_sizes, int n_in,
                              void* d_out, int out_size, void* d_ws, size_t ws_size,
                              hipStream_t stream) {
    const float* x  = (const float*)d_in[0];
    const float* pe = (const float*)d_in[1];
    float* out      = (float*)d_out;
    (void)in_sizes; (void)n_in; (void)out_size; (void)d_ws; (void)ws_size;

    dim3 grid(B_ * F_);
    dim3 block(THREADS);
    hipLaunchKernelGGL(pe2d_add_kernel, grid, block, 0, stream, x, pe, out);
}